// SurrogateEncoder_7078106104245
// MI455X (gfx1250) — compile-verified
//
#include <hip/hip_runtime.h>

// ---------------------------------------------------------------------------
// SurrogateEncoder for MI455X (gfx1250, wave32, WMMA)
//   x      = sum_t emb[seq[:,t]]                        [N,256]
//   deg    = segsum(1, dst) + 1 ; dis = rsqrt(deg)
//   layer:  h = x@W + b ; agg = segsum(h[src]*dis_s*dis_d, dst) + h*dis^2
//   out    = segsum(relu(layer2), batch)                [64,256]
// ---------------------------------------------------------------------------

#define N_NODES 50000
#define N_EDGES 800000
#define N_GRAPHS 64
#define DIM 256

typedef __attribute__((ext_vector_type(2))) float v2f;
typedef __attribute__((ext_vector_type(8))) float v8f;

// ---------------------------------------------------------------- utilities
__global__ void zero_kernel(float* __restrict__ p, int n) {
    int i = blockIdx.x * blockDim.x + threadIdx.x;
    if (i < n) p[i] = 0.0f;
}

// --------------------------------------------------------- embedding gather
// one block per node: 256 threads, one output dim each; tokens staged in LDS
__global__ __launch_bounds__(256) void embed_kernel(const int* __restrict__ seq,
                                                    const float* __restrict__ emb,
                                                    float* __restrict__ x) {
    int node = blockIdx.x;
    int d = threadIdx.x;
    __shared__ int toks[16];
    if (threadIdx.x < 16) toks[threadIdx.x] = seq[node * 16 + threadIdx.x];
    __syncthreads();
    float s = 0.0f;
#pragma unroll
    for (int t = 0; t < 16; ++t)
        s += emb[(size_t)toks[t] * DIM + d];
    x[(size_t)node * DIM + d] = s;
}

// ------------------------------------------------------------------ degrees
__global__ void degree_kernel(const int* __restrict__ dst, float* __restrict__ deg,
                              int e_cnt) {
    int e = blockIdx.x * blockDim.x + threadIdx.x;
    if (e < e_cnt) atomicAdd(&deg[dst[e]], 1.0f);
}

__global__ void dis_kernel(const float* __restrict__ deg, float* __restrict__ dis,
                           int n) {
    int i = blockIdx.x * blockDim.x + threadIdx.x;
    if (i < n) dis[i] = rsqrtf(deg[i] + 1.0f);   // self-loop included
}

// ------------------------------------------------------------- WMMA GEMM
// out[N,256] = (relu_a ? relu(A) : A)[N,256] @ W[256,256] + bias
// block: 256 threads = 8 waves; block tile 32(M) x 64(N); wave tile 16x16.
// K staged through LDS in 64-wide chunks; 16 v_wmma_f32_16x16x4_f32 per chunk.
__global__ __launch_bounds__(256) void gemm_wmma_kernel(const float* __restrict__ A,
                                                        const float* __restrict__ W,
                                                        const float* __restrict__ bias,
                                                        float* __restrict__ out,
                                                        int n_rows, int relu_a) {
    __shared__ float Xs[32 * 64];   // A tile  [32 rows][64 k]
    __shared__ float Ws[64 * 64];   // W tile  [64 k][64 cols]

    const int tid  = threadIdx.x;
    const int wave = tid >> 5;
    const int lane = tid & 31;
    const int wm   = wave >> 2;       // 0..1  : wave row within block tile
    const int wn   = wave & 3;        // 0..3  : wave col within block tile
    const int hi   = lane >> 4;       // lane half (ISA 16x16 layout)
    const int ml   = lane & 15;

    const int m0 = blockIdx.x * 32;
    const int n0 = blockIdx.y * 64;

    v8f acc = {};

    for (int kc = 0; kc < DIM; kc += 64) {
        // ---- stage A tile (guarded, relu-fused) -------------------------
#pragma unroll
        for (int i = 0; i < 8; ++i) {
            int idx = tid + i * 256;          // 2048 elements
            int r = idx >> 6, c = idx & 63;
            int row = m0 + r;
            float v = (row < n_rows) ? A[(size_t)row * DIM + kc + c] : 0.0f;
            if (relu_a) v = fmaxf(v, 0.0f);
            Xs[idx] = v;
        }
        // ---- stage W tile ----------------------------------------------
#pragma unroll
        for (int i = 0; i < 16; ++i) {
            int idx = tid + i * 256;          // 4096 elements
            int r = idx >> 6, c = idx & 63;
            Ws[idx] = W[(size_t)(kc + r) * DIM + n0 + c];
        }
        __syncthreads();

        // ---- 16 WMMAs over this K chunk --------------------------------
#pragma unroll
        for (int k0 = 0; k0 < 64; k0 += 4) {
            // A 16x4 fragment: VGPR0 = K=k0 (+2 for hi half), VGPR1 = K+1
            const float* xr = &Xs[(wm * 16 + ml) * 64 + k0 + 2 * hi];
            v2f a; a.x = xr[0]; a.y = xr[1];
            // B 4x16 fragment: row-striped across lanes per VGPR
            v2f b;
            b.x = Ws[(k0 + 2 * hi) * 64 + wn * 16 + ml];
            b.y = Ws[(k0 + 2 * hi + 1) * 64 + wn * 16 + ml];
            acc = __builtin_amdgcn_wmma_f32_16x16x4_f32(
                /*neg_a=*/false, a, /*neg_b=*/false, b,
                /*c_mod=*/(short)0, acc, /*reuse_a=*/false, /*reuse_b=*/false);
        }
        __syncthreads();
    }

    // ---- epilogue: bias add, guarded store ------------------------------
    const int col = n0 + wn * 16 + ml;
    const float bv = bias[col];
#pragma unroll
    for (int v = 0; v < 8; ++v) {
        int row = m0 + wm * 16 + v + 8 * hi;   // C layout: VGPR v -> M = v (+8)
        if (row < n_rows)
            out[(size_t)row * DIM + col] = acc[v] + bv;
    }
}

// --------------------------------------------- self-loop init of aggregate
// agg[i,:] = h[i,:] * dis[i]^2   (full overwrite -> no pre-zero needed)
__global__ __launch_bounds__(256) void selfloop_kernel(const float* __restrict__ h,
                                                       const float* __restrict__ dis,
                                                       float* __restrict__ agg) {
    int node = blockIdx.x;
    int d = threadIdx.x;
    float s = dis[node];
    agg[(size_t)node * DIM + d] = h[(size_t)node * DIM + d] * s * s;
}

// ----------------------------------------------------------- edge scatter
// agg[dst,:] += h[src,:] * dis[src]*dis[dst]   (GLOBAL_ATOMIC_ADD_F32)
// one wave per edge, 8 coalesced dims per lane
__global__ __launch_bounds__(256) void scatter_kernel(const int* __restrict__ src,
                                                      const int* __restrict__ dst,
                                                      const float* __restrict__ dis,
                                                      const float* __restrict__ h,
                                                      float* __restrict__ agg,
                                                      int e_cnt) {
    int e = blockIdx.x * 8 + (threadIdx.x >> 5);
    if (e >= e_cnt) return;
    int lane = threadIdx.x & 31;
    int s = src[e];
    int d = dst[e];
    float norm = dis[s] * dis[d];
    const float* hs = h + (size_t)s * DIM;
    float* ad = agg + (size_t)d * DIM;
#pragma unroll
    for (int j = 0; j < 8; ++j) {
        int c = lane + j * 32;
        atomicAdd(&ad[c], hs[c] * norm);
    }
}

// ------------------------------------------------------- relu + graph pool
__global__ __launch_bounds__(256) void pool_kernel(const float* __restrict__ agg,
                                                   const int* __restrict__ batch,
                                                   float* __restrict__ out) {
    int node = blockIdx.x;
    int d = threadIdx.x;
    int g = batch[node];
    float v = fmaxf(agg[(size_t)node * DIM + d], 0.0f);
    atomicAdd(&out[(size_t)g * DIM + d], v);
}

// ---------------------------------------------------------------------------
extern "C" void kernel_launch(void* const* d_in, const int* in_sizes, int n_in,
                              void* d_out, int out_size, void* d_ws, size_t ws_size,
                              hipStream_t stream) {
    (void)in_sizes; (void)n_in; (void)out_size; (void)ws_size;

    const int*   seq   = (const int*)d_in[0];
    const int*   edge  = (const int*)d_in[1];   // [2, E] flat
    const int*   batch = (const int*)d_in[2];
    const float* emb   = (const float*)d_in[3];
    const float* W0    = (const float*)d_in[4];
    const float* b0    = (const float*)d_in[5];
    const float* W1    = (const float*)d_in[6];
    const float* b1    = (const float*)d_in[7];
    float*       out   = (float*)d_out;

    const int* src = edge;
    const int* dst = edge + N_EDGES;

    // workspace layout (floats): buf0 | buf1 | deg | dis   (~102.8 MB)
    float* buf0 = (float*)d_ws;
    float* buf1 = buf0 + (size_t)N_NODES * DIM;
    float* deg  = buf1 + (size_t)N_NODES * DIM;
    float* dis  = deg + N_NODES;

    const dim3 blk(256);
    const dim3 gemm_grid((N_NODES + 31) / 32, DIM / 64);

    // degree: zero -> count -> rsqrt
    zero_kernel<<<(N_NODES + 255) / 256, blk, 0, stream>>>(deg, N_NODES);
    degree_kernel<<<(N_EDGES + 255) / 256, blk, 0, stream>>>(dst, deg, N_EDGES);
    dis_kernel<<<(N_NODES + 255) / 256, blk, 0, stream>>>(deg, dis, N_NODES);

    // statement encoder: x = sum_t emb[seq]  -> buf0
    embed_kernel<<<N_NODES, blk, 0, stream>>>(seq, emb, buf0);

    // layer 0: h = x@W0 + b0 -> buf1 ; agg -> buf0
    gemm_wmma_kernel<<<gemm_grid, blk, 0, stream>>>(buf0, W0, b0, buf1, N_NODES, 0);
    selfloop_kernel<<<N_NODES, blk, 0, stream>>>(buf1, dis, buf0);
    scatter_kernel<<<(N_EDGES + 7) / 8, blk, 0, stream>>>(src, dst, dis, buf1, buf0,
                                                          N_EDGES);

    // layer 1: h = relu(agg)@W1 + b1 -> buf1 ; agg -> buf0
    gemm_wmma_kernel<<<gemm_grid, blk, 0, stream>>>(buf0, W1, b1, buf1, N_NODES, 1);
    selfloop_kernel<<<N_NODES, blk, 0, stream>>>(buf1, dis, buf0);
    scatter_kernel<<<(N_EDGES + 7) / 8, blk, 0, stream>>>(src, dst, dis, buf1, buf0,
                                                          N_EDGES);

    // pooled = segsum(relu(agg), batch)
    zero_kernel<<<(N_GRAPHS * DIM + 255) / 256, blk, 0, stream>>>(out, N_GRAPHS * DIM);
    pool_kernel<<<N_NODES, blk, 0, stream>>>(buf0, batch, out);
}